// FeatureGATCausal_1090921693404
// MI455X (gfx1250) — compile-verified
//
#include <hip/hip_runtime.h>
#include <hip/hip_bf16.h>

// ---------------------------------------------------------------------------
// FeatureGATCausal forward for MI455X (gfx1250, wave32).
// fp32 end-to-end; matmuls on V_WMMA_F32_16X16X4_F32 (full-precision matrix
// pipe — workload is ~6 GFLOP / ~40 MB, L2-resident, latency bound, so low
// precision buys nothing and fp32 WMMA matches the fp32 reference).
// GEMM tile staging uses GLOBAL_LOAD_ASYNC_TO_LDS_B128 (ASYNCcnt-tracked
// direct global->LDS DMA) when the toolchain exposes the builtin; otherwise
// falls back to the global_load_b128 + ds_store path.
// ---------------------------------------------------------------------------

typedef float v2f __attribute__((ext_vector_type(2)));
typedef float v8f __attribute__((ext_vector_type(8)));

#if defined(__gfx1250__)
#if __has_builtin(__builtin_amdgcn_global_load_async_to_lds_b128) && \
    __has_builtin(__builtin_amdgcn_s_wait_asynccnt)
#define USE_ASYNC_LDS 1
#endif
#endif
#ifndef USE_ASYNC_LDS
#define USE_ASYNC_LDS 0
#endif

#if USE_ASYNC_LDS
// Builtin signature (from clang diagnostic):
//   void __builtin_amdgcn_global_load_async_to_lds_b128(
//       int4 AS(1)* gaddr, int4 AS(3)* ldsaddr, imm int offset, imm int cpol)
typedef int v4i_b128 __attribute__((vector_size(16)));
typedef __attribute__((address_space(1))) v4i_b128 as1_v4i;
typedef __attribute__((address_space(3))) v4i_b128 as3_v4i;
#endif

// ---------------------------------------------------------------------------
// Kernel 0: pack the four [64,1024] inputs into contiguous x[256,1024]
// ---------------------------------------------------------------------------
__global__ __launch_bounds__(256) void pack_x(
    const float* __restrict__ a, const float* __restrict__ b,
    const float* __restrict__ c, const float* __restrict__ d,
    float* __restrict__ x) {
  int idx = blockIdx.x * 256 + threadIdx.x;      // 0 .. 262143
  int seg = idx >> 16;                           // 64*1024 = 65536 per segment
  int off = idx & 65535;
  const float* src = (seg == 0) ? a : (seg == 1) ? b : (seg == 2) ? c : d;
  x[idx] = src[off];
}

// ---------------------------------------------------------------------------
// Kernel 1: C[256,N] = A[256,K] @ W[K,N] + bias[N]   (fp32 WMMA 16x16x4)
// Block = 128 threads (4 waves). Block tile = 16(M) x 64(N); wave w owns the
// 16x16 tile at nBlock + 16*w. K staged through LDS in chunks of 32.
// ---------------------------------------------------------------------------
__global__ __launch_bounds__(128) void gemm_wmma_f32(
    const float* __restrict__ A, const float* __restrict__ W,
    const float* __restrict__ bias, float* __restrict__ C,
    int K, int N) {
  __shared__ float As[16][33];   // +1 pad: frag reads walk rows (stride 33)
  __shared__ float Bs[32][66];   // +2 pad: decorrelate the two half-wave rows

  const int t      = threadIdx.x;
  const int wave   = t >> 5;
  const int lane   = t & 31;
  const int l15    = lane & 15;
  const int koff   = (lane >> 4) << 1;       // 0 for lanes 0-15, 2 for 16-31
  const int mBase  = blockIdx.y << 4;
  const int nBlock = blockIdx.x << 6;
  const int nCol   = nBlock + (wave << 4);

  v8f acc = {};

  for (int k0 = 0; k0 < K; k0 += 32) {
#if USE_ASYNC_LDS
    // ---- async DMA staging: direct global->LDS, tracked by ASYNCcnt ----
    {
      int id = t << 2;                       // A tile: one b128 per thread
      int r = id >> 5, cc = id & 31;
      __builtin_amdgcn_global_load_async_to_lds_b128(
          (as1_v4i*)(A + (mBase + r) * K + k0 + cc),
          (as3_v4i*)&As[r][cc], 0, 0);
    }
    #pragma unroll
    for (int e = 0; e < 4; ++e) {            // B tile: four b128 per thread
      int id = (e << 7) + t;
      int r = id >> 4, c4 = (id & 15) << 2;
      __builtin_amdgcn_global_load_async_to_lds_b128(
          (as1_v4i*)(W + (k0 + r) * N + nBlock + c4),
          (as3_v4i*)&Bs[r][c4], 0, 0);
    }
    __builtin_amdgcn_s_wait_asynccnt(0);
#else
    // ---- fallback: load to VGPR, store to LDS ----
    {
      int id = t << 2;                       // 0..508
      int r = id >> 5, cc = id & 31;
      const float4 av = *(const float4*)(A + (mBase + r) * K + k0 + cc);
      As[r][cc + 0] = av.x; As[r][cc + 1] = av.y;
      As[r][cc + 2] = av.z; As[r][cc + 3] = av.w;
    }
    #pragma unroll
    for (int e = 0; e < 4; ++e) {
      int id = (e << 7) + t;                 // float4 index 0..511
      int r = id >> 4, c4 = (id & 15) << 2;
      const float4 bv = *(const float4*)(W + (k0 + r) * N + nBlock + c4);
      Bs[r][c4 + 0] = bv.x; Bs[r][c4 + 1] = bv.y;
      Bs[r][c4 + 2] = bv.z; Bs[r][c4 + 3] = bv.w;
      if (e == 0 && k0 + 32 < K)             // hint next B tile into cache
        __builtin_prefetch((const void*)(W + (k0 + 32 + r) * N + nBlock + c4), 0, 3);
    }
#endif
    __syncthreads();

    // ---- 8 x V_WMMA_F32_16X16X4_F32 over this K chunk ----
    #pragma unroll
    for (int kk = 0; kk < 32; kk += 4) {
      v2f a, b;
      // A 16x4 frag: M = l15, K = koff / koff+1 (ISA 7.12.2, 32-bit A)
      a.x = As[l15][kk + koff];
      a.y = As[l15][kk + koff + 1];
      // B 4x16 frag: K = koff / koff+1, N = l15
      b.x = Bs[kk + koff    ][(wave << 4) + l15];
      b.y = Bs[kk + koff + 1][(wave << 4) + l15];
      acc = __builtin_amdgcn_wmma_f32_16x16x4_f32(
          /*neg_a=*/false, a, /*neg_b=*/false, b,
          /*c_mod=*/(short)0, acc, /*reuse_a=*/false, /*reuse_b=*/false);
    }
    __syncthreads();
  }

  // ---- epilogue: C/D layout — VGPR r: M = r + 8*(lane>=16), N = l15 ----
  const int rowOff = (lane >> 4) << 3;
  #pragma unroll
  for (int r = 0; r < 8; ++r) {
    int row = mBase + r + rowOff;
    int col = nCol + l15;
    C[row * N + col] = acc[r] + bias[col];
  }
}

// ---------------------------------------------------------------------------
// Kernel 2: GATv2 attention (dense graph, no self-loops), generic over Dh.
// One 256-thread block per (target i, head h). N = 256, H = 4.
//   s[j] = sum_d leaky_relu(xr[i,h,d] + xl[j,h,d]) * att[h,d]
//   a = softmax_j(s, s[i] = -inf)
//   out[i,h,d] = (sum_j a[j] * xl[j,h,d]) + bias[h*Dh+d]   (+ ELU if layer 1)
// ---------------------------------------------------------------------------
__global__ __launch_bounds__(256) void gat_attn(
    const float* __restrict__ xl, const float* __restrict__ xr,
    const float* __restrict__ att, const float* __restrict__ bias,
    float* __restrict__ out, int Dh, int applyElu) {
  const int N = 256;
  const int i = blockIdx.x >> 2;
  const int h = blockIdx.x & 3;
  const int stride = 4 * Dh;                 // H * Dh
  const int t = threadIdx.x;

  __shared__ float xr_sh[512];
  __shared__ float att_sh[512];
  __shared__ float sc[256];
  __shared__ float red[256];

  for (int d = t; d < Dh; d += 256) {
    xr_sh[d]  = xr[i * stride + h * Dh + d];
    att_sh[d] = att[h * Dh + d];
  }
  __syncthreads();

  // --- scores: thread t handles source j = t ---
  {
    const float* xlj = xl + t * stride + h * Dh;
    float s = 0.f;
    for (int d = 0; d < Dh; ++d) {
      float v = xr_sh[d] + xlj[d];
      v = (v > 0.f) ? v : 0.2f * v;          // leaky_relu(0.2)
      s += v * att_sh[d];
    }
    sc[t] = (t == i) ? -__builtin_inff() : s;   // mask self-loop
  }
  __syncthreads();

  // --- softmax over j (block reduction) ---
  red[t] = sc[t];
  __syncthreads();
  for (int off = 128; off > 0; off >>= 1) {
    if (t < off) red[t] = fmaxf(red[t], red[t + off]);
    __syncthreads();
  }
  float mx = red[0];
  __syncthreads();
  float ex = __expf(sc[t] - mx);
  sc[t] = ex;                                 // unnormalized weight
  red[t] = ex;
  __syncthreads();
  for (int off = 128; off > 0; off >>= 1) {
    if (t < off) red[t] += red[t + off];
    __syncthreads();
  }
  const float inv = 1.f / red[0];

  // --- aggregation: out[i,h,d] = inv * sum_j sc[j]*xl[j,h,d] + bias ---
  for (int d = t; d < Dh; d += 256) {
    float accv = 0.f;
    for (int j = 0; j < N; ++j)
      accv += sc[j] * xl[j * stride + h * Dh + d];
    accv = accv * inv + bias[h * Dh + d];
    if (applyElu) accv = (accv > 0.f) ? accv : (__expf(accv) - 1.f);
    out[i * stride + h * Dh + d] = accv;
  }
}

// ---------------------------------------------------------------------------
// Kernel 3: per-row LayerNorm + ReLU + residual; writes the 128 output rows.
// Residual rows come directly from video_1_fea / video_2_fea.
// ---------------------------------------------------------------------------
__global__ __launch_bounds__(256) void ln_relu_res(
    const float* __restrict__ y, const float* __restrict__ g,
    const float* __restrict__ bb, const float* __restrict__ v1,
    const float* __restrict__ v2, const float* __restrict__ rw,
    float* __restrict__ out) {
  const int C = 1024;
  const int r = blockIdx.x;                  // 0..127
  const int t = threadIdx.x;
  const float* yr  = y + r * C;
  const float* res = (r < 64) ? (v1 + r * C) : (v2 + (r - 64) * C);
  __shared__ float red[256];

  float s = 0.f;
  for (int c = t; c < C; c += 256) s += yr[c];
  red[t] = s;
  __syncthreads();
  for (int off = 128; off > 0; off >>= 1) {
    if (t < off) red[t] += red[t + off];
    __syncthreads();
  }
  const float mu = red[0] * (1.f / C);
  __syncthreads();

  float vs = 0.f;
  for (int c = t; c < C; c += 256) { float dd = yr[c] - mu; vs += dd * dd; }
  red[t] = vs;
  __syncthreads();
  for (int off = 128; off > 0; off >>= 1) {
    if (t < off) red[t] += red[t + off];
    __syncthreads();
  }
  const float inv = rsqrtf(red[0] * (1.f / C) + 1e-5f);
  const float w = rw[0];

  for (int c = t; c < C; c += 256) {
    float v = (yr[c] - mu) * inv * g[c] + bb[c];
    v = fmaxf(v, 0.f);
    out[r * C + c] = v + w * res[c];
  }
}

// ---------------------------------------------------------------------------
// Launch: pack -> gemm(xl1) -> gemm(xr1) -> attn1(+elu) -> gemm(xl2) ->
//         gemm(xr2) -> attn2 -> gemm(y) -> layernorm/relu/residual
// Workspace (floats, with reuse; peak 1,835,008 floats = ~7.3 MB):
//   X    @ 0         [262144]   (dead after gemm1/2 -> reused by XL2)
//   XL1  @ 262144    [524288]   (dead after attn1   -> reused by Y)
//   XR1  @ 786432    [524288]   (dead after attn1   -> reused by XR2, OUT2)
//   OUT1 @ 1310720   [524288]
// ---------------------------------------------------------------------------
extern "C" void kernel_launch(void* const* d_in, const int* in_sizes, int n_in,
                              void* d_out, int out_size, void* d_ws, size_t ws_size,
                              hipStream_t stream) {
  const float* v1    = (const float*)d_in[0];
  const float* v2    = (const float*)d_in[1];
  const float* f1    = (const float*)d_in[2];
  const float* f2    = (const float*)d_in[3];
  const float* Wl1   = (const float*)d_in[4];
  const float* bl1   = (const float*)d_in[5];
  const float* Wr1   = (const float*)d_in[6];
  const float* br1   = (const float*)d_in[7];
  const float* att1  = (const float*)d_in[8];
  const float* bias1 = (const float*)d_in[9];
  const float* Wl2   = (const float*)d_in[10];
  const float* bl2   = (const float*)d_in[11];
  const float* Wr2   = (const float*)d_in[12];
  const float* br2   = (const float*)d_in[13];
  const float* att2  = (const float*)d_in[14];
  const float* bias2 = (const float*)d_in[15];
  const float* Wout  = (const float*)d_in[16];
  const float* bout  = (const float*)d_in[17];
  const float* ln_g  = (const float*)d_in[18];
  const float* ln_b  = (const float*)d_in[19];
  const float* rw    = (const float*)d_in[20];

  float* ws   = (float*)d_ws;
  float* X    = ws;                 // 256*1024
  float* XL1  = ws + 262144;        // 256*2048
  float* XR1  = ws + 786432;        // 256*2048
  float* OUT1 = ws + 1310720;       // 256*2048
  float* XL2  = ws;                 // 256*1024 (reuse X)
  float* XR2  = ws + 786432;        // 256*1024 (reuse XR1 lo)
  float* OUT2 = ws + 1048576;       // 256*1024 (reuse XR1 hi)
  float* Y    = ws + 262144;        // 256*1024 (reuse XL1 lo)

  pack_x<<<1024, 256, 0, stream>>>(v1, v2, f1, f2, X);

  // GAT layer 1: K=1024, N=2048
  gemm_wmma_f32<<<dim3(32, 16), 128, 0, stream>>>(X, Wl1, bl1, XL1, 1024, 2048);
  gemm_wmma_f32<<<dim3(32, 16), 128, 0, stream>>>(X, Wr1, br1, XR1, 1024, 2048);
  gat_attn<<<1024, 256, 0, stream>>>(XL1, XR1, att1, bias1, OUT1, 512, /*elu=*/1);

  // GAT layer 2: K=2048, N=1024
  gemm_wmma_f32<<<dim3(16, 16), 128, 0, stream>>>(OUT1, Wl2, bl2, XL2, 2048, 1024);
  gemm_wmma_f32<<<dim3(16, 16), 128, 0, stream>>>(OUT1, Wr2, br2, XR2, 2048, 1024);
  gat_attn<<<1024, 256, 0, stream>>>(XL2, XR2, att2, bias2, OUT2, 256, /*elu=*/0);

  // Output projection + LayerNorm + ReLU + residual (only 128 rows emitted)
  gemm_wmma_f32<<<dim3(16, 16), 128, 0, stream>>>(OUT2, Wout, bout, Y, 1024, 1024);
  ln_relu_res<<<128, 256, 0, stream>>>(Y, ln_g, ln_b, v1, v2, rw, (float*)d_out);
}